// ContinuousBinaryTreeConvLayer_43885975830917
// MI455X (gfx1250) — compile-verified
//
#include <hip/hip_runtime.h>

// ContinuousBinaryTreeConv: out[bn,o] = relu( sum_k sum_f agg[bn,k,f]*w[k][f,o] + b[o] )
// agg_t = node vec; agg_r/agg_l = coefficient-weighted child sums.
// GEMM M=65536, K=384, N=128 on V_WMMA_F32_16X16X4_F32 (fp32 matrix pipe).
// W is repacked once per launch into per-lane fragment order so the steady-state
// loop is: 1x global_load_b128 (B, 2 k-steps) + 1x ds_load_2addr_b64 (A) + 2x WMMA.

typedef float v2f __attribute__((ext_vector_type(2)));
typedef float v8f __attribute__((ext_vector_type(8)));

#define BATCH   32
#define NNODES  2048
#define NCH     8
#define FDIM    128
#define ODIM    128
#define KTOT    384
#define LSTRIDE 388   // 384 + 4-float pad: rows step 4 LDS banks -> conflict-free frag loads

#define NCTILE  8                      // 128 / 16 output column tiles
#define NS2     48                     // 384 / 8 : two k-steps per packed float4
#define PACK_FLOAT4 (NCTILE * NS2 * 32)           // 12288 float4 = 196608 B
#define PACK_BYTES  (PACK_FLOAT4 * 16)

__device__ __forceinline__ void fma4(float4& a, float s, const float4 v) {
    a.x += s * v.x; a.y += s * v.y; a.z += s * v.z; a.w += s * v.w;
}

// ---- repack stacked W = [w_t; w_r; w_l] (384x128) into WMMA B-fragment order ----
// packed[((c*48 + s2)*32 + lane)] = { W[k0+2h][col], W[k0+2h+1][col],
//                                     W[k0+4+2h][col], W[k0+4+2h+1][col] }
// with k0 = 8*s2, h = lane>>4, col = c*16 + (lane&15).
__global__ __launch_bounds__(256)
void pack_w_kernel(const float* __restrict__ wt, const float* __restrict__ wr,
                   const float* __restrict__ wl, float* __restrict__ packed)
{
    const int idx  = blockIdx.x * 256 + threadIdx.x;   // 0 .. PACK_FLOAT4-1
    const int lane = idx & 31;
    const int grp  = idx >> 5;
    const int s2   = grp % NS2;
    const int c    = grp / NS2;
    const int half = lane >> 4;
    const int l16  = lane & 15;
    const int col  = c * 16 + l16;
    const int r0   = s2 * 8 + 2 * half;       // rows for first WMMA of the pair
    const int r2   = r0 + 4;                  // rows for second WMMA of the pair

    auto elem = [&](int kk) -> float {
        const float* w = (kk < 128) ? wt : (kk < 256) ? wr : wl;
        return w[(size_t)(kk & 127) * ODIM + col];
    };
    float4 v = { elem(r0), elem(r0 + 1), elem(r2), elem(r2 + 1) };
    ((float4*)packed)[idx] = v;
}

template <bool PACKED>
__global__ __launch_bounds__(256)
void ctbconv_kernel(const float* __restrict__ nodes,
                    const int*   __restrict__ children,
                    const float* __restrict__ w_t,
                    const float* __restrict__ w_l,
                    const float* __restrict__ w_r,
                    const float* __restrict__ bias,
                    const float* __restrict__ packed,
                    float*       __restrict__ out)
{
    __shared__ float lds[16 * LSTRIDE];

    const int tid       = threadIdx.x;
    const int node_base = blockIdx.x * 16;          // flat b*N + n, tile of 16 nodes

    // ---------------- Phase 1: per-node aggregation into LDS A-panel [16 x 384] ----
    {
        const int m    = tid >> 4;                  // node within tile (0..15)
        const int fi   = (tid & 15) * 8;            // this thread's 8 features
        const int node = node_base + m;
        const int bb   = node / NNODES;

        const float4* nv = (const float4*)(nodes + (size_t)node * FDIM + fi);
        float4 t0 = nv[0], t1 = nv[1];

        const int* chp = children + (size_t)node * NCH;
        int c[NCH];
        int nz = 0;
        #pragma unroll
        for (int j = 0; j < NCH; ++j) { c[j] = chp[j]; nz += (c[j] != 0); }
        const bool  is_single = (nz == 1);
        const float inv_denom = is_single ? 1.0f : 1.0f / (float)(nz - 1);

        float4 r0 = {0,0,0,0}, r1 = {0,0,0,0};
        float4 l0 = {0,0,0,0}, l1 = {0,0,0,0};
        #pragma unroll
        for (int j = 0; j < NCH; ++j) {
            const int sj = j + 1;                   // slot index 1..8
            const int cj = c[j];
            if (cj != 0) {                          // cj==0 -> zero vector contributes nothing
                const float cr = is_single ? ((sj == 1) ? 0.5f : 0.0f)
                                           : (float)(sj - 1) * inv_denom;
                const float cl = 1.0f - cr;
                const float4* cv =
                    (const float4*)(nodes + ((size_t)bb * NNODES + cj) * FDIM + fi);
                float4 v0 = cv[0], v1 = cv[1];
                fma4(r0, cr, v0); fma4(r1, cr, v1);
                fma4(l0, cl, v0); fma4(l1, cl, v1);
            }
        }

        float* row = lds + m * LSTRIDE;
        *(float4*)(row +   0 + fi) = t0;  *(float4*)(row +   0 + fi + 4) = t1;
        *(float4*)(row + 128 + fi) = r0;  *(float4*)(row + 128 + fi + 4) = r1;
        *(float4*)(row + 256 + fi) = l0;  *(float4*)(row + 256 + fi + 4) = l1;
    }

    __syncthreads();

    // ---------------- Phase 2: WMMA GEMM, one wave per 16-column output tile ------
    const int wv   = tid >> 5;                      // wave -> output column tile 0..7
    const int lane = tid & 31;
    const int half = lane >> 4;                     // 0: K+0/K+1, 1: K+2/K+3
    const int l16  = lane & 15;
    const int col  = wv * 16 + l16;

    v8f acc = {};
    const float* la = lds + l16 * LSTRIDE + 2 * half;

    if constexpr (PACKED) {
        const float4* wp = (const float4*)packed + (size_t)wv * NS2 * 32 + lane;
        #pragma unroll
        for (int s2 = 0; s2 < NS2; ++s2) {
            const float4 b4 = wp[(size_t)s2 * 32];  // one b128: B frags for 2 k-steps
            const int k0 = s2 * 8;
            v2f a0 = *(const v2f*)(la + k0);
            v2f a1 = *(const v2f*)(la + k0 + 4);
            v2f b0; b0.x = b4.x; b0.y = b4.y;
            v2f b1; b1.x = b4.z; b1.y = b4.w;
            acc = __builtin_amdgcn_wmma_f32_16x16x4_f32(
                      false, a0, false, b0, (short)0, acc, false, false);
            acc = __builtin_amdgcn_wmma_f32_16x16x4_f32(
                      false, a1, false, b1, (short)0, acc, false, false);
        }
    } else {
        const float* wsegs[3] = { w_t, w_r, w_l };  // K-stack order matches coef stack
        #pragma unroll
        for (int seg = 0; seg < 3; ++seg) {
            const float* ws = wsegs[seg] + (size_t)(2 * half) * ODIM + col;
            const float* ls = la + seg * FDIM;
            #pragma unroll
            for (int s = 0; s < 32; ++s) {
                const int k0 = 4 * s;
                v2f a = *(const v2f*)(ls + k0);
                v2f bf;
                bf.x = ws[(size_t)k0 * ODIM];
                bf.y = ws[(size_t)k0 * ODIM + ODIM];
                acc = __builtin_amdgcn_wmma_f32_16x16x4_f32(
                          false, a, false, bf, (short)0, acc, false, false);
            }
        }
    }

    // ---------------- Epilogue: bias + relu + store ------------------------------
    const float bv = bias[col];
    float* op = out + (size_t)node_base * ODIM + col;
    #pragma unroll
    for (int r = 0; r < 8; ++r) {
        const int mrow = half ? (r + 8) : r;        // D: VGPR r = rows r / r+8
        float v = acc[r] + bv;
        op[(size_t)mrow * ODIM] = v > 0.0f ? v : 0.0f;
    }
}

extern "C" void kernel_launch(void* const* d_in, const int* in_sizes, int n_in,
                              void* d_out, int out_size, void* d_ws, size_t ws_size,
                              hipStream_t stream) {
    // setup_inputs order: nodes, children, w_t, w_l, w_r, b
    const float* nodes    = (const float*)d_in[0];
    const int*   children = (const int*)  d_in[1];
    const float* w_t      = (const float*)d_in[2];
    const float* w_l      = (const float*)d_in[3];
    const float* w_r      = (const float*)d_in[4];
    const float* bias     = (const float*)d_in[5];
    float*       outp     = (float*)d_out;

    const dim3 grid((BATCH * NNODES) / 16);

    if (ws_size >= (size_t)PACK_BYTES) {
        float* packed = (float*)d_ws;
        pack_w_kernel<<<PACK_FLOAT4 / 256, 256, 0, stream>>>(w_t, w_r, w_l, packed);
        ctbconv_kernel<true><<<grid, 256, 0, stream>>>(
            nodes, children, w_t, w_l, w_r, bias, packed, outp);
    } else {
        ctbconv_kernel<false><<<grid, 256, 0, stream>>>(
            nodes, children, w_t, w_l, w_r, bias, nullptr, outp);
    }
}